// HistogramLoss_876173328933
// MI455X (gfx1250) — compile-verified
//
#include <hip/hip_runtime.h>
#include <hip/hip_bf16.h>
#include <stdint.h>

// ---------------------------------------------------------------------------
// HistogramLoss on MI455X (gfx1250, wave32)
//
// reference == per-(b,c) exact histogram matching (rank matching, tie-free):
//   matched[argsort(ref*sm)[i]] = sort(tgt*tm)[i];  loss = mean((src*sm - matched*sm)^2)
// Two 8x4-bit LSD radix sorts per channel (stable scatter with packed
// Hillis-Steele rank scan), then a gather + squared-diff reduction.
// CDNA5 paths: V_WMMA_F32_16X16X4_F32 wave reduction (full f32 precision),
// TENSOR_LOAD_TO_LDS (TDM) streaming + s_wait_tensorcnt, global_prefetch_b8.
// ---------------------------------------------------------------------------

#define BATCH   16
#define CHN     3
#define NCH     (BATCH*CHN)      // 48 channels
#define NPC     (512*512)        // 262144 elements per channel
#define TOT     (NCH*NPC)        // 12,582,912
#define RADIX   16
#define PASSES  8
#define TPB     256
#define EPT     16
#define TILE    (TPB*EPT)        // 4096
#define TILES   (NPC/TILE)       // 64

typedef __attribute__((ext_vector_type(2))) float v2f;
typedef __attribute__((ext_vector_type(8))) float v8f;
typedef unsigned __attribute__((ext_vector_type(4))) v4u;
typedef int __attribute__((ext_vector_type(4))) v4i;
typedef int __attribute__((ext_vector_type(8))) v8i;

#if defined(__has_builtin)
#if __has_builtin(__builtin_amdgcn_tensor_load_to_lds) && \
    __has_builtin(__builtin_amdgcn_s_wait_tensorcnt)
#define HAVE_TDM 1
#endif
#endif

__device__ __forceinline__ unsigned keyOf(float f) {
  unsigned u = __float_as_uint(f);
  return (u & 0x80000000u) ? ~u : (u | 0x80000000u);
}
__device__ __forceinline__ float unKey(unsigned k) {
  unsigned u = (k & 0x80000000u) ? (k ^ 0x80000000u) : ~k;
  return __uint_as_float(u);
}

#ifdef HAVE_TDM
// Issue a 1-D TDM copy of `ndw` dwords (ndw <= 65535) global -> LDS.
// D# group0: count=1 | lds_addr | global_addr | type=2 ("image").
// D# group1: data_size=4B, tensor_dim0 = tile_dim0 = ndw, dims 1+ unused.
// clang-23 / therock-10.0 builtin form (6 args):
//   (uint32x4 g0, int32x8 g1, int32x4 g2, int32x4 g3, int32x8 extra, i32 cpol)
__device__ __forceinline__ void tdm_load_dwords(unsigned lds_off,
                                                const void* gptr,
                                                unsigned ndw) {
  unsigned long long ga = (unsigned long long)gptr;
  v4u g0;
  g0[0] = 1u;                                                  // count=1
  g0[1] = lds_off;                                             // lds_addr
  g0[2] = (unsigned)ga;                                        // gaddr[31:0]
  g0[3] = (unsigned)((ga >> 32) & 0x01ffffffu) | 0x80000000u;  // [56:32]|type2
  v8i g1;
  g1[0] = (int)(2u << 16);                   // data_size = 2 -> 4 bytes
  g1[1] = (int)((ndw & 0xffffu) << 16);      // tensor_dim0[15:0] in bits 63:48
  g1[2] = (int)((ndw >> 16) | (1u << 16));   // tensor_dim0[31:16] | tensor_dim1=1
  g1[3] = (int)(ndw << 16);                  // tile_dim0 = ndw (bits 127:112)
  g1[4] = 1;                                 // tile_dim1 = 1
  g1[5] = (int)ndw;                          // tensor_dim0_stride[31:0]
  g1[6] = 0;
  g1[7] = 0;
  v4i z4 = {0, 0, 0, 0};
  v8i z8 = {0, 0, 0, 0, 0, 0, 0, 0};
  __builtin_amdgcn_tensor_load_to_lds(g0, g1, z4, z4, z8, 0);
}
#endif

// ---- build transformed radix keys ------------------------------------------
__global__ void build_ref_k(const float* __restrict__ ref_img,
                            const float* __restrict__ src_mask,
                            unsigned* __restrict__ keys,
                            unsigned* __restrict__ idx) {
  int g = blockIdx.x * TPB + threadIdx.x;
  int ch = g / NPC;
  int pix = g - ch * NPC;
  int b = ch / CHN;
  float sm = src_mask[(size_t)b * NPC + pix];
  float v = (ref_img[g] * 0.5f + 0.5f) * 255.0f * sm;
  keys[g] = keyOf(v);
  idx[g]  = (unsigned)pix;
}

__global__ void build_tgt_k(const float* __restrict__ tgt_img,
                            const float* __restrict__ tgt_mask,
                            unsigned* __restrict__ keys) {
  int g = blockIdx.x * TPB + threadIdx.x;
  int ch = g / NPC;
  int pix = g - ch * NPC;
  int b = ch / CHN;
  float tm = tgt_mask[(size_t)b * NPC + pix];
  float v = (tgt_img[g] * 0.5f + 0.5f) * 255.0f * tm;
  keys[g] = keyOf(v);
}

// ---- radix pass: per-tile histogram ----------------------------------------
__global__ void hist_k(const unsigned* __restrict__ keys,
                       unsigned* __restrict__ hist, int shift) {
  __shared__ unsigned lh[RADIX];
  int t = threadIdx.x, tile = blockIdx.x, ch = blockIdx.y;
  if (t < RADIX) lh[t] = 0u;
  __syncthreads();
  size_t base = (size_t)ch * NPC + (size_t)tile * TILE + (size_t)t * EPT;
  if (tile + 1 < TILES)
    __builtin_prefetch(keys + base + TILE, 0, 0);   // global_prefetch_b8
  const uint4* p4 = reinterpret_cast<const uint4*>(keys + base);
#pragma unroll
  for (int q = 0; q < EPT / 4; ++q) {
    uint4 v = p4[q];
    atomicAdd(&lh[(v.x >> shift) & 15u], 1u);
    atomicAdd(&lh[(v.y >> shift) & 15u], 1u);
    atomicAdd(&lh[(v.z >> shift) & 15u], 1u);
    atomicAdd(&lh[(v.w >> shift) & 15u], 1u);
  }
  __syncthreads();
  if (t < RADIX) hist[((size_t)ch * RADIX + t) * TILES + tile] = lh[t];
}

// ---- radix pass: per-channel scan of (digit, tile) counts -> global bases --
__global__ void scan_k(unsigned* __restrict__ hist) {
  __shared__ unsigned tot[RADIX];
  int ch = blockIdx.x, d = threadIdx.x;
  unsigned* h = hist + (size_t)ch * RADIX * TILES;
  if (d < RADIX) {
    unsigned run = 0;
    for (int tl = 0; tl < TILES; ++tl) {
      unsigned c = h[d * TILES + tl];
      h[d * TILES + tl] = run;
      run += c;
    }
    tot[d] = run;
  }
  __syncthreads();
  if (d == 0) {
    unsigned run = 0;
    for (int i = 0; i < RADIX; ++i) { unsigned c = tot[i]; tot[i] = run; run += c; }
  }
  __syncthreads();
  if (d < RADIX) {
    unsigned b = tot[d];
    for (int tl = 0; tl < TILES; ++tl) h[d * TILES + tl] += b;
  }
}

// ---- radix pass: stable scatter --------------------------------------------
// Cross-thread stable ranks via packed 16x16-bit Hillis-Steele scan:
// per-thread digit counts <= 16, tile prefix <= 4096 < 2^16 -> no carry
// between packed fields; 8 parallel steps with all 256 threads active.
__global__ void scatter_k(const unsigned* __restrict__ inK,
                          const unsigned* __restrict__ inP,
                          unsigned* __restrict__ outK,
                          unsigned* __restrict__ outP,
                          const unsigned* __restrict__ hist,
                          int shift, int hasPayload) {
  __shared__ unsigned rankm[RADIX][TPB];       // 16 KB
  __shared__ unsigned scanw[RADIX / 2][TPB];   // 8 KB (packed scan)
  __shared__ unsigned gbase[RADIX];
  int t = threadIdx.x, tile = blockIdx.x, ch = blockIdx.y;
  size_t chbase = (size_t)ch * NPC;
  size_t base = chbase + (size_t)tile * TILE + (size_t)t * EPT;

  unsigned k[EPT], pld[EPT];
  const uint4* p4 = reinterpret_cast<const uint4*>(inK + base);
#pragma unroll
  for (int q = 0; q < EPT / 4; ++q) {
    uint4 v = p4[q];
    k[q*4+0]=v.x; k[q*4+1]=v.y; k[q*4+2]=v.z; k[q*4+3]=v.w;
  }
  if (hasPayload) {
    const uint4* q4 = reinterpret_cast<const uint4*>(inP + base);
#pragma unroll
    for (int q = 0; q < EPT / 4; ++q) {
      uint4 v = q4[q];
      pld[q*4+0]=v.x; pld[q*4+1]=v.y; pld[q*4+2]=v.z; pld[q*4+3]=v.w;
    }
  }
#pragma unroll
  for (int d = 0; d < RADIX; ++d) rankm[d][t] = 0u;
  if (t < RADIX) gbase[t] = hist[((size_t)ch * RADIX + t) * TILES + tile];
  __syncthreads();
  // per-thread digit counts (thread owns column t -> no races)
#pragma unroll
  for (int e = 0; e < EPT; ++e) {
    unsigned d = (k[e] >> shift) & (RADIX - 1);
    rankm[d][t] += 1u;
  }
  __syncthreads();
  // pack 16 counters into 8 words of two 16-bit fields
  unsigned own[RADIX / 2];
#pragma unroll
  for (int w = 0; w < RADIX / 2; ++w) {
    own[w] = rankm[2 * w][t] | (rankm[2 * w + 1][t] << 16);
    scanw[w][t] = own[w];
  }
  __syncthreads();
  // inclusive Hillis-Steele scan over threads (8 steps)
  for (int step = 1; step < TPB; step <<= 1) {
    unsigned add[RADIX / 2];
#pragma unroll
    for (int w = 0; w < RADIX / 2; ++w)
      add[w] = (t >= step) ? scanw[w][t - step] : 0u;
    __syncthreads();
#pragma unroll
    for (int w = 0; w < RADIX / 2; ++w) scanw[w][t] += add[w];
    __syncthreads();
  }
  // exclusive cross-thread offsets back into rankm[d][t]
#pragma unroll
  for (int w = 0; w < RADIX / 2; ++w) {
    unsigned excl = scanw[w][t] - own[w];
    rankm[2 * w][t]     = excl & 0xffffu;
    rankm[2 * w + 1][t] = excl >> 16;
  }
  __syncthreads();
  // replay in-order: pos = global base + cross-thread prefix + within-thread
#pragma unroll
  for (int e = 0; e < EPT; ++e) {
    unsigned d = (k[e] >> shift) & (RADIX - 1);
    unsigned pos = gbase[d] + rankm[d][t];
    rankm[d][t] += 1u;
    outK[chbase + pos] = k[e];
    if (hasPayload) outP[chbase + pos] = pld[e];
  }
}

// ---- stage-1 reduction: TDM-streamed gather + WMMA f32 wave reduce ---------
__global__ void reduce1_k(const unsigned* __restrict__ tgtKeys,   // sorted
                          const unsigned* __restrict__ refIdx,    // argsort perm
                          const float* __restrict__ src_img,
                          const float* __restrict__ src_mask,
                          float* __restrict__ partials) {
  int t = threadIdx.x, tile = blockIdx.x, ch = blockIdx.y;
  int b = ch / CHN;
  size_t chbase = (size_t)ch * NPC;
#ifdef HAVE_TDM
  __shared__ unsigned ldsK[TILE];   // 16 KB
  __shared__ unsigned ldsI[TILE];   // 16 KB
  if (threadIdx.x < 32) {           // wave 0 drives the Tensor Data Mover
    tdm_load_dwords((unsigned)(uintptr_t)ldsK,
                    tgtKeys + chbase + (size_t)tile * TILE, TILE);
    tdm_load_dwords((unsigned)(uintptr_t)ldsI,
                    refIdx + chbase + (size_t)tile * TILE, TILE);
    __builtin_amdgcn_s_wait_tensorcnt(0);
  }
  __syncthreads();
#endif
  float a0 = 0.0f, a1 = 0.0f;
#pragma unroll
  for (int e = 0; e < EPT; ++e) {
    int sl = e * TPB + t;                 // position within tile (conflict-free)
#ifdef HAVE_TDM
    float tv = unKey(ldsK[sl]);           // tgt_sorted[s]
    unsigned j = ldsI[sl];                // pixel with rank s
#else
    size_t s = chbase + (size_t)tile * TILE + sl;
    float tv = unKey(tgtKeys[s]);
    unsigned j = refIdx[s];
#endif
    float sm = src_mask[(size_t)b * NPC + j];
    float sv = (src_img[chbase + j] * 0.5f + 0.5f) * 255.0f * sm;
    float d = sv - tv * sm;
    if (e & 1) a1 += d * d; else a0 += d * d;
  }
  // Wave (32-lane) reduction via V_WMMA_F32_16X16X4_F32 with ones B-matrix:
  // A[16x4] holds the wave's 64 accumulators; D rows = per-row sums.
  v2f a;    a.x = a0;   a.y = a1;
  v2f ones; ones.x = 1.0f; ones.y = 1.0f;
  v8f c = {};
  v8f dm = __builtin_amdgcn_wmma_f32_16x16x4_f32(
      /*neg_a=*/false, a, /*neg_b=*/false, ones,
      /*c_mod=*/(short)0, c, /*reuse_a=*/false, /*reuse_b=*/false);
  float s = dm[0] + dm[1] + dm[2] + dm[3] + dm[4] + dm[5] + dm[6] + dm[7];
  s += __shfl_xor(s, 16, 32);                       // rows 0-7 + rows 8-15
  __shared__ float wsum[TPB / 32];
  int wave = t >> 5;
  if ((t & 31) == 0) wsum[wave] = s;
  __syncthreads();
  if (t == 0) {
    float tot = 0.0f;
    for (int w = 0; w < TPB / 32; ++w) tot += wsum[w];  // fixed order
    partials[(size_t)blockIdx.y * gridDim.x + blockIdx.x] = tot;
  }
}

// ---- stage-2 reduction: deterministic tree ---------------------------------
__global__ void reduce2_k(const float* __restrict__ partials,
                          float* __restrict__ out) {
  __shared__ float sh[1024];
  int t = threadIdx.x;
  float s = 0.0f;
  for (int i = t; i < NCH * TILES; i += 1024) s += partials[i];
  sh[t] = s;
  __syncthreads();
  for (int off = 512; off > 0; off >>= 1) {
    if (t < off) sh[t] += sh[t + off];
    __syncthreads();
  }
  if (t == 0) out[0] = sh[0] * (1.0f / (float)TOT);
}

// ---------------------------------------------------------------------------
extern "C" void kernel_launch(void* const* d_in, const int* in_sizes, int n_in,
                              void* d_out, int out_size, void* d_ws, size_t ws_size,
                              hipStream_t stream) {
  const float* src_img     = (const float*)d_in[0];
  const float* target_img  = (const float*)d_in[1];
  const float* src_mask    = (const float*)d_in[2];
  const float* target_mask = (const float*)d_in[3];
  const float* ref_img     = (const float*)d_in[4];
  float* out = (float*)d_out;

  // workspace layout: 4x TOT u32 ping-pong buffers + hist + partials (~192 MB)
  unsigned* K0 = (unsigned*)d_ws;       // ref keys A  / tgt ping
  unsigned* K1 = K0 + (size_t)TOT;      // ref keys B  / tgt pong (final tgt)
  unsigned* P0 = K1 + (size_t)TOT;      // ref payload A (final perm)
  unsigned* P1 = P0 + (size_t)TOT;      // ref payload B
  unsigned* hist = P1 + (size_t)TOT;    // NCH*RADIX*TILES u32
  float* partials = (float*)(hist + (size_t)NCH * RADIX * TILES);

  dim3 gtiles(TILES, NCH);

  // --- argsort(ref * sm): keys K0<->K1, payload P0<->P1; ends in K0/P0 ---
  build_ref_k<<<TOT / TPB, TPB, 0, stream>>>(ref_img, src_mask, K0, P0);
  for (int p = 0; p < PASSES; ++p) {
    unsigned* ik = (p & 1) ? K1 : K0;
    unsigned* ok = (p & 1) ? K0 : K1;
    unsigned* ip = (p & 1) ? P1 : P0;
    unsigned* op = (p & 1) ? P0 : P1;
    hist_k   <<<gtiles, TPB, 0, stream>>>(ik, hist, p * 4);
    scan_k   <<<NCH, 32, 0, stream>>>(hist);
    scatter_k<<<gtiles, TPB, 0, stream>>>(ik, ip, ok, op, hist, p * 4, 1);
  }
  // final ref permutation now in P0 (keys in K0, dead)

  // --- sort(tgt * tm) keys only: start K1, ping-pong with K0; ends in K1 ---
  build_tgt_k<<<TOT / TPB, TPB, 0, stream>>>(target_img, target_mask, K1);
  for (int p = 0; p < PASSES; ++p) {
    unsigned* ik = (p & 1) ? K0 : K1;
    unsigned* ok = (p & 1) ? K1 : K0;
    hist_k   <<<gtiles, TPB, 0, stream>>>(ik, hist, p * 4);
    scan_k   <<<NCH, 32, 0, stream>>>(hist);
    scatter_k<<<gtiles, TPB, 0, stream>>>(ik, P1, ok, P1, hist, p * 4, 0);
  }

  // --- loss = mean((src*sm - tgt_sorted[rank]*sm)^2) ---
  reduce1_k<<<gtiles, TPB, 0, stream>>>(K1, P0, src_img, src_mask, partials);
  reduce2_k<<<1, 1024, 0, stream>>>(partials, out);
  (void)in_sizes; (void)n_in; (void)out_size; (void)ws_size;
}